// Conv2DTransposeSeparableLayer_55980603736310
// MI455X (gfx1250) — compile-verified
//
#include <hip/hip_runtime.h>

// CDNA5 / gfx1250: wave32, WMMA f32 16x16x4.
typedef __attribute__((ext_vector_type(2))) float v2f;
typedef __attribute__((ext_vector_type(4))) float v4f;
typedef __attribute__((ext_vector_type(8))) float v8f;

// Fixed inverse wavelet filters: [band][tap]
__device__ __constant__ float c_inv[3][9] = {
  { 0.0f, 0.0f, 1.0f/16.0f, 0.5f, 14.0f/16.0f, 0.5f, 1.0f/16.0f, 0.0f, 0.0f },
  { -1.0f/128.0f, -1.0f/16.0f, -10.0f/64.0f, -7.0f/16.0f, 85.0f/64.0f,
    -7.0f/16.0f, -10.0f/64.0f, -1.0f/16.0f, -1.0f/128.0f },
  { 1.0f/256.0f, 1.0f/32.0f, 15.0f/128.0f, 17.0f/32.0f, 0.0f,
    -17.0f/32.0f, -15.0f/128.0f, -1.0f/32.0f, -1.0f/256.0f }
};

// Constant banded weight matrix W[K=(p,ic), e]: 12 padded source lines x 3
// bands -> 16 outputs of a tile (e = 2q + r, stride-2 transposed conv).
__device__ __forceinline__ float wfun(int K, int e) {
  int p = K / 3, ic = K - 3 * p;
  int q = e >> 1, r = e & 1;
  int u = p - q;
  bool valid = r ? (u >= 1 && u <= 4) : (u >= 0 && u <= 4);
  int t = 8 + r - 2 * u;
  t = t < 0 ? 0 : (t > 8 ? 8 : t);
  return valid ? c_inv[ic][t] : 0.0f;
}

// Symmetric-pad reflection for length-256 axis, pad 3: jp in [0,261].
__device__ __forceinline__ int refl(int jp) {
  return (jp < 3) ? (2 - jp) : ((jp > 258) ? (514 - jp) : (jp - 3));
}
__device__ __forceinline__ bool is_border(int jp) { return (jp < 3) || (jp > 258); }

// ---------------------------------------------------------------------------
// Pass X: in [16,256,256,36] -> mid [16,256,512,12]
// Tile: D[M=group(12/16), N=16 x-outputs] = A(16x36 gathered) x B(36x16 const)
// Border sign + reflection folded into B / the load offsets (both h-invariant);
// inner loop = 18 saddr loads + 9 WMMA + 3 float4 stores.
// ---------------------------------------------------------------------------
__global__ __launch_bounds__(256) void invx_wmma(const float* __restrict__ in,
                                                 float* __restrict__ mid) {
  const int lane = threadIdx.x & 31;
  const int wv = __builtin_amdgcn_readfirstlane(
      (int)(blockIdx.x * 256 + threadIdx.x) >> 5);
  const int hb = wv & 31;           // 32 h-blocks of 8 rows
  const int xt = (wv >> 5) & 31;    // 32 x-tiles of 16 outputs
  const int b  = wv >> 10;          // 16 batches
  const int hl = lane >> 4;         // half-wave K-pair select
  const int e  = lane & 15;         // M for A / N for B,D
  const int m0 = xt << 3;
  const int x0 = xt << 4;
  const int gM = e < 12 ? e : 11;   // clamp A row (rows 12..15 discarded)

  int offs[9][2];                   // h-invariant per-lane load offsets
  v2f Bw[9];                        // constant weights, border sign folded in
#pragma unroll
  for (int kc = 0; kc < 9; ++kc) {
#pragma unroll
    for (int v = 0; v < 2; ++v) {
      int K = 4 * kc + 2 * hl + v;
      int p = K / 3, ic = K - 3 * p;
      int jp = m0 + 1 + p;          // padded column in [1,260]
      offs[kc][v] = refl(jp) * 36 + gM * 3 + ic;
      float w = wfun(K, e);
      if (ic == 2 && is_border(jp)) w = -w;   // fold odd-band border sign
      if (v == 0) Bw[kc].x = w; else Bw[kc].y = w;
    }
  }
  const int soff = e * 12;
  const float* inrow = in + (size_t)(b * 256 + hb * 8) * (256 * 36);
  float* orow = mid + ((size_t)(b * 256 + hb * 8) * 512 + x0) * 12;

  for (int hi = 0; hi < 8; ++hi) {
    v8f acc = {};
#pragma unroll
    for (int kc = 0; kc < 9; ++kc) {
      v2f a;
      a.x = inrow[offs[kc][0]];
      a.y = inrow[offs[kc][1]];
      acc = __builtin_amdgcn_wmma_f32_16x16x4_f32(
          false, a, false, Bw[kc], (short)0, acc, false, false);
    }
    // D: VGPR d -> M = d + 8*hl (group), lane -> N = e. Rows M<12 stored,
    // contiguous in memory -> aligned float4 stores (48B lane stride).
    if (hl == 0) {
      *(v4f*)(orow + soff)     = v4f{acc[0], acc[1], acc[2], acc[3]};
      *(v4f*)(orow + soff + 4) = v4f{acc[4], acc[5], acc[6], acc[7]};
    } else {
      *(v4f*)(orow + soff + 8) = v4f{acc[0], acc[1], acc[2], acc[3]};
    }
    inrow += 256 * 36;
    orow  += 512 * 12;
  }
}

// ---------------------------------------------------------------------------
// Pass Y: mid [16,256,512,12] -> out [16,512,512,4]
// Tile: D[M=16 y-outputs, N=16 x] = A(16x36 const) x B(36x16 gathered)
// Interior y-tiles (30 of 32): saddr loads with precomputed offsets, no
// reflect/sign work. Border tiles take the scalar-branch slow path.
// ---------------------------------------------------------------------------
__global__ __launch_bounds__(256) void invy_wmma(const float* __restrict__ mid,
                                                 float* __restrict__ out) {
  const int lane = threadIdx.x & 31;
  const int wv = __builtin_amdgcn_readfirstlane(
      (int)(blockIdx.x * 256 + threadIdx.x) >> 5);
  const int yb = wv & 3;            // 4 blocks of 8 y-tiles
  const int xt = (wv >> 2) & 31;    // 32 x-tiles
  const int g  = (wv >> 7) & 3;     // 4 output groups
  const int b  = wv >> 9;           // 16 batches
  const int hl = lane >> 4;
  const int e  = lane & 15;
  const int x0 = xt << 4;
  const int x  = x0 + e;

  v2f Aw[9];                        // constant weight operand (reused)
  int poffs[9][2];                  // interior offsets rel. to (n0-2) row base
#pragma unroll
  for (int kc = 0; kc < 9; ++kc) {
#pragma unroll
    for (int v = 0; v < 2; ++v) {
      int K = 4 * kc + 2 * hl + v;
      int p = K / 3, ic = K - 3 * p;
      poffs[kc][v] = p * 6144 + x * 12 + ic;   // 6144 = 512*12
      float w = wfun(K, e);
      if (v == 0) Aw[kc].x = w; else Aw[kc].y = w;
    }
  }
  const float* ub = mid + (size_t)b * (256 * 512 * 12) + 3 * g;
  const int yoff = hl * (8 * 512 * 4) + e * 4;

  for (int yi = 0; yi < 8; ++yi) {
    int yt = yb * 8 + yi;
    int n0 = yt << 3;
    int y0 = yt << 4;
    v8f acc = {};
    if (n0 >= 2 && n0 <= 246) {     // interior: no reflection, no sign
      const float* ybase = ub + (size_t)(n0 - 2) * 6144;
#pragma unroll
      for (int kc = 0; kc < 9; ++kc) {
        v2f bb;
        bb.x = ybase[poffs[kc][0]];
        bb.y = ybase[poffs[kc][1]];
        acc = __builtin_amdgcn_wmma_f32_16x16x4_f32(
            false, Aw[kc], false, bb, (short)0, acc, false, false);
      }
    } else {                        // border: reflect + odd-band sign (rare)
#pragma unroll
      for (int kc = 0; kc < 9; ++kc) {
        v2f bb;
#pragma unroll
        for (int v = 0; v < 2; ++v) {
          int K = 4 * kc + 2 * hl + v;
          int p = K / 3, ic = K - 3 * p;
          int jp = n0 + 1 + p;      // padded row in [1,260]
          float s = (ic == 2 && is_border(jp)) ? -1.0f : 1.0f;
          float val = ub[(size_t)refl(jp) * 6144 + x * 12 + ic] * s;
          if (v == 0) bb.x = val; else bb.y = val;
        }
        acc = __builtin_amdgcn_wmma_f32_16x16x4_f32(
            false, Aw[kc], false, bb, (short)0, acc, false, false);
      }
    }
    // D: VGPR d -> M = d + 8*hl (y offset), lane -> N = e (x offset).
    float* ob = out + ((size_t)(b * 512 + y0) * 512 + x0) * 4 + g;
#pragma unroll
    for (int d = 0; d < 8; ++d)
      ob[yoff + d * 2048] = acc[d];
  }
}

extern "C" void kernel_launch(void* const* d_in, const int* in_sizes, int n_in,
                              void* d_out, int out_size, void* d_ws, size_t ws_size,
                              hipStream_t stream) {
  (void)in_sizes; (void)n_in; (void)out_size; (void)ws_size;
  const float* in = (const float*)d_in[0];
  float* out = (float*)d_out;
  float* mid = (float*)d_ws;   // 16*256*512*12 floats = 100.7 MB (L2-resident)

  // Pass X: 16*32*32 = 16384 waves -> 2048 blocks of 256 (8 waves each)
  invx_wmma<<<2048, 256, 0, stream>>>(in, mid);
  // Pass Y: 16*4*32*4 = 8192 waves -> 1024 blocks of 256
  invy_wmma<<<1024, 256, 0, stream>>>(mid, out);
}